// _SpanMarker_26671746908734
// MI455X (gfx1250) — compile-verified
//
#include <hip/hip_runtime.h>

typedef __attribute__((ext_vector_type(2))) float v2f;
typedef __attribute__((ext_vector_type(4))) float v4f;
typedef __attribute__((ext_vector_type(8))) float v8f;

#define B_    8
#define L_    512
#define D_    768
#define MAXW_ 12
#define N_    (L_ * MAXW_)   // 6144 spans per batch
#define ROWS1 (B_ * L_)      // 4096 rows for the projection GEMMs
#define ROWS2 (B_ * N_)      // 49152 span rows for the output GEMM
#define PITCH 772            // 768 + 4 pad: conflict-free ds_load_b64 across wave32

// ---------------------------------------------------------------------------
// Weight pre-pack: P[(k/2)*D + n] = { W[k][n], W[k+1][n] }  (one v2f each)
// Turns the per-WMMA B-fragment fetch into a single coalesced global_load_b64.
// ---------------------------------------------------------------------------
__launch_bounds__(256)
__global__ void k_pack_pairs(const float* __restrict__ W,
                             float* __restrict__ P, int K) {
    const int idx   = blockIdx.x * 256 + threadIdx.x;   // over (K/2)*D_
    const int total = (K >> 1) * D_;
    if (idx < total) {
        const int k2 = idx / D_;
        const int n  = idx - k2 * D_;
        v2f p;
        p.x = W[(size_t)(2 * k2) * D_ + n];
        p.y = W[(size_t)(2 * k2 + 1) * D_ + n];
        ((v2f*)P)[idx] = p;
    }
}

// ---------------------------------------------------------------------------
// Stage 1: sr = h @ Ws + bs ; er = h @ We + be   (blockIdx.z selects which)
// grid = (D_/256, ROWS1/16, 2), block = 256 (8 waves)
// 16 M-rows staged in LDS; 8 waves x 2 accumulators cover 256 N-cols.
// B comes from the K-paired packed weight (v2f per fragment half).
// ---------------------------------------------------------------------------
__launch_bounds__(256)
__global__ void k_proj_f32wmma(const float* __restrict__ h,
                               const float* __restrict__ Wsp,
                               const float* __restrict__ bs,
                               const float* __restrict__ Wep,
                               const float* __restrict__ be,
                               float* __restrict__ sr,
                               float* __restrict__ er) {
    __shared__ float lds_a[16 * PITCH];

    const int    tid = threadIdx.x;
    const v2f*   Wp  = (const v2f*)(blockIdx.z ? Wep : Wsp);
    const float* bia = blockIdx.z ? be : bs;
    float*       out = blockIdx.z ? er : sr;
    const int    m0  = blockIdx.y * 16;

    // Cooperative load of the 16 x 768 A-panel into LDS (float4 per op).
    {
        const float* src = h + (size_t)m0 * D_;
        for (int i = tid; i < 16 * (D_ / 4); i += 256) {
            const int row = i / (D_ / 4);
            const int c4  = (i - row * (D_ / 4)) * 4;
            v4f v = *(const v4f*)(src + (size_t)row * D_ + c4);
            *(v4f*)(&lds_a[row * PITCH + c4]) = v;
        }
    }
    __syncthreads();

    const int wave = tid >> 5;
    const int lane = tid & 31;
    const int m    = lane & 15;            // A-matrix row this lane carries
    const int koff = (lane >> 4) << 1;     // lanes 16-31 carry K+2/K+3
    const int kh   = lane >> 4;            // packed-pair row offset (0 or 1)
    const int nc   = lane & 15;            // B/C column within tile
    const int n0   = blockIdx.x * 256 + wave * 32;

    v8f c0 = {};
    v8f c1 = {};
    const float* lrow  = &lds_a[m * PITCH + koff];
    const v2f*   brow  = Wp + (size_t)kh * D_ + n0 + nc;

#pragma unroll 4
    for (int kk = 0; kk < D_; kk += 4) {
        v2f a = *(const v2f*)(lrow + kk);                       // ds_load_b64
        const v2f* bp = brow + (size_t)(kk >> 1) * D_;
        v2f b0 = bp[0];                                         // global_load_b64
        v2f b1 = bp[16];                                        // global_load_b64
        c0 = __builtin_amdgcn_wmma_f32_16x16x4_f32(false, a, false, b0,
                                                   (short)0, c0, false, false);
        c1 = __builtin_amdgcn_wmma_f32_16x16x4_f32(false, a, false, b1,
                                                   (short)0, c1, false, false);
    }

    // C/D layout: VGPR v -> row (v + 8*(lane>=16)), column = lane&15.
    const int   mbase = (lane >> 4) * 8;
    const float bb0 = bia[n0 + nc];
    const float bb1 = bia[n0 + 16 + nc];
#pragma unroll
    for (int v = 0; v < 8; ++v) {
        const size_t row = (size_t)(m0 + mbase + v) * D_;
        out[row + n0 + nc]      = c0[v] + bb0;
        out[row + n0 + 16 + nc] = c1[v] + bb1;
    }
}

// ---------------------------------------------------------------------------
// Stage 2: out[r] = relu(sr[rs[r]]) @ Wo[0:768] + relu(er[re[r]]) @ Wo[768:1536] + bo
// grid = (D_/256, ROWS2/16), block = 256 (8 waves)
// Gathered+ReLU'd rows staged in LDS, two K=768 WMMA passes reusing one panel.
// ---------------------------------------------------------------------------
__launch_bounds__(256)
__global__ void k_span_f32wmma(const long long* __restrict__ span_idx,
                               const float* __restrict__ sr,
                               const float* __restrict__ er,
                               const float* __restrict__ Wop,
                               const float* __restrict__ bo,
                               float* __restrict__ out) {
    __shared__ float lds_a[16 * PITCH];
    __shared__ int   rs[16];
    __shared__ int   re[16];

    const int tid = threadIdx.x;
    const int r0  = blockIdx.y * 16;

    if (tid < 16) {
        const int r = r0 + tid;               // flat span row in [0, ROWS2)
        const int b = r / N_;
        const long long* p = span_idx + (size_t)r * 2;
        rs[tid] = b * L_ + (int)p[0];
        re[tid] = b * L_ + (int)p[1];
    }
    __syncthreads();

    const int wave = tid >> 5;
    const int lane = tid & 31;
    const int m    = lane & 15;
    const int koff = (lane >> 4) << 1;
    const int kh   = lane >> 4;
    const int nc   = lane & 15;
    const int n0   = blockIdx.x * 256 + wave * 32;

    v8f c0 = {};
    v8f c1 = {};
    const float* lrow = &lds_a[m * PITCH + koff];

#pragma unroll
    for (int phase = 0; phase < 2; ++phase) {
        // Gather 16 rows (start reps, then end reps) into LDS with ReLU.
        const float* src  = phase ? er : sr;
        const int*   ridx = phase ? re : rs;
        for (int i = tid; i < 16 * (D_ / 4); i += 256) {
            const int row = i / (D_ / 4);
            const int c4  = (i - row * (D_ / 4)) * 4;
            v4f v = *(const v4f*)(src + (size_t)ridx[row] * D_ + c4);
            v.x = fmaxf(v.x, 0.0f);
            v.y = fmaxf(v.y, 0.0f);
            v.z = fmaxf(v.z, 0.0f);
            v.w = fmaxf(v.w, 0.0f);
            *(v4f*)(&lds_a[row * PITCH + c4]) = v;
        }
        __syncthreads();

        // Packed Wo: 768 pair-rows total; phase 1 starts at pair-row 384.
        const v2f* brow = (const v2f*)Wop
                        + (size_t)(phase * (D_ / 2) + kh) * D_ + n0 + nc;
#pragma unroll 4
        for (int kk = 0; kk < D_; kk += 4) {
            v2f a = *(const v2f*)(lrow + kk);
            const v2f* bp = brow + (size_t)(kk >> 1) * D_;
            v2f b0 = bp[0];
            v2f b1 = bp[16];
            c0 = __builtin_amdgcn_wmma_f32_16x16x4_f32(false, a, false, b0,
                                                       (short)0, c0, false, false);
            c1 = __builtin_amdgcn_wmma_f32_16x16x4_f32(false, a, false, b1,
                                                       (short)0, c1, false, false);
        }
        __syncthreads();   // panel will be overwritten next phase
    }

    const int   mbase = (lane >> 4) * 8;
    const float bb0 = bo[n0 + nc];
    const float bb1 = bo[n0 + 16 + nc];
#pragma unroll
    for (int v = 0; v < 8; ++v) {
        const size_t row = (size_t)(r0 + mbase + v) * D_;
        out[row + n0 + nc]      = c0[v] + bb0;
        out[row + n0 + 16 + nc] = c1[v] + bb1;
    }
}

// ---------------------------------------------------------------------------
extern "C" void kernel_launch(void* const* d_in, const int* in_sizes, int n_in,
                              void* d_out, int out_size, void* d_ws, size_t ws_size,
                              hipStream_t stream) {
    (void)in_sizes; (void)n_in; (void)out_size; (void)ws_size;

    const float*     h    = (const float*)d_in[0];
    const long long* span = (const long long*)d_in[1];
    const float*     Ws   = (const float*)d_in[2];
    const float*     bs   = (const float*)d_in[3];
    const float*     We   = (const float*)d_in[4];
    const float*     be   = (const float*)d_in[5];
    const float*     Wo   = (const float*)d_in[6];
    const float*     bo   = (const float*)d_in[7];
    float*           out  = (float*)d_out;

    // Workspace layout (floats):
    //   sr   : ROWS1*D_          (12.6 MB)
    //   er   : ROWS1*D_          (12.6 MB)
    //   Ws_p : D_*D_             ( 2.4 MB)  K-paired pack of Ws
    //   We_p : D_*D_             ( 2.4 MB)  K-paired pack of We
    //   Wo_p : 2*D_*D_           ( 4.7 MB)  K-paired pack of Wo
    float* sr  = (float*)d_ws;
    float* er  = sr  + (size_t)ROWS1 * D_;
    float* Wsp = er  + (size_t)ROWS1 * D_;
    float* Wep = Wsp + (size_t)D_ * D_;
    float* Wop = Wep + (size_t)D_ * D_;

    // One-shot weight packs (streaming, ~19 MB total traffic).
    {
        const int tD  = (D_ / 2) * D_;        // 294912 pairs
        const int t2D = D_ * D_;              // 589824 pairs (Wo, K=1536)
        k_pack_pairs<<<(tD  + 255) / 256, 256, 0, stream>>>(Ws, Wsp, D_);
        k_pack_pairs<<<(tD  + 255) / 256, 256, 0, stream>>>(We, Wep, D_);
        k_pack_pairs<<<(t2D + 255) / 256, 256, 0, stream>>>(Wo, Wop, 2 * D_);
    }

    dim3 g1(D_ / 256, ROWS1 / 16, 2);               // (3, 256, 2)
    k_proj_f32wmma<<<g1, 256, 0, stream>>>(h, Wsp, bs, Wep, be, sr, er);

    dim3 g2(D_ / 256, ROWS2 / 16, 1);               // (3, 3072)
    k_span_f32wmma<<<g2, 256, 0, stream>>>(span, sr, er, Wop, bo, out);
}